// CADI_53609781788982
// MI455X (gfx1250) — compile-verified
//
#include <hip/hip_runtime.h>
#include <hip/hip_bf16.h>

// N=50000 nodes, E=1600000 edges, H=64
// d_in order: x[ N*64 ], edge_index[2*E], edge_attr[E*64], attn[E],
//             nn_w[64*64], nn_b[64], wr_w[64], wr_b[1], eps[1]
// d_out: [ N*64 floats (nn(out)) | E floats (w_rel) ]
// d_ws : agg buffer, N*64 floats

typedef __attribute__((ext_vector_type(2))) float v2f;
typedef __attribute__((ext_vector_type(8))) float v8f;

#define H_DIM 64

// Guaranteed single-instruction non-returning fp32 atomic add at device scope.
// (Avoids clang's CAS-loop expansion of atomicrmw fadd.)
__device__ __forceinline__ void atomic_add_f32_dev(float* p, float v) {
  asm volatile("global_atomic_add_f32 %0, %1, off scope:SCOPE_DEV"
               :
               : "v"(p), "v"(v)
               : "memory");
}

// ---------------------------------------------------------------------------
// Kernel 1: agg = (1 + eps) * x          (vectorized float4)
// ---------------------------------------------------------------------------
__global__ void cadi_init(const float* __restrict__ x,
                          const float* __restrict__ eps,
                          float* __restrict__ agg, int n4) {
  int i = blockIdx.x * blockDim.x + threadIdx.x;
  if (i >= n4) return;
  const float s = 1.0f + eps[0];
  float4 v = ((const float4*)x)[i];
  v.x *= s; v.y *= s; v.z *= s; v.w *= s;
  ((float4*)agg)[i] = v;
}

// ---------------------------------------------------------------------------
// Kernel 2: per-edge message + scatter-add.
// 16 lanes per edge, each lane owns 4 contiguous floats (float4).
// ---------------------------------------------------------------------------
__global__ void cadi_edge(const float* __restrict__ x,
                          const int* __restrict__ src,
                          const int* __restrict__ dst,
                          const float* __restrict__ edge_attr,
                          const float* __restrict__ attn,
                          const float* __restrict__ wr_w,
                          const float* __restrict__ wr_b,
                          float* __restrict__ agg,
                          float* __restrict__ wrel_out,
                          int E) {
  __shared__ float s_wr[H_DIM];
  __shared__ float s_wrb;
  if (threadIdx.x < H_DIM) s_wr[threadIdx.x] = wr_w[threadIdx.x];
  if (threadIdx.x == 0) s_wrb = wr_b[0];
  __syncthreads();

  const int tid = blockIdx.x * blockDim.x + threadIdx.x;
  const int e = tid >> 4;        // 16 threads per edge
  const int l = tid & 15;        // lane-in-edge
  if (e >= E) return;

  const float4 ea = *(const float4*)(edge_attr + (size_t)e * H_DIM + l * 4);
  const float4 w4 = *(const float4*)(s_wr + l * 4);

  // per-lane partial of edge_attr[e,:] . wr_w
  float p = ea.x * w4.x + ea.y * w4.y + ea.z * w4.z + ea.w * w4.w;
  // reduce across the 16 lanes of this edge (wave32: two edges per wave)
  p += __shfl_xor(p, 8, 16);
  p += __shfl_xor(p, 4, 16);
  p += __shfl_xor(p, 2, 16);
  p += __shfl_xor(p, 1, 16);
  const float wrel = p + s_wrb;

  const int sn = src[e];
  const int dn = dst[e];
  const float a = attn[e];
  const float4 xj = *(const float4*)(x + (size_t)sn * H_DIM + l * 4);

  float4 m;
  m.x = fmaxf(0.0f, xj.x * a + wrel * ea.x);
  m.y = fmaxf(0.0f, xj.y * a + wrel * ea.y);
  m.z = fmaxf(0.0f, xj.z * a + wrel * ea.z);
  m.w = fmaxf(0.0f, xj.w * a + wrel * ea.w);

  float* ap = agg + (size_t)dn * H_DIM + l * 4;
  // native global_atomic_add_f32 into the L2-resident 12.8MB agg buffer
  atomic_add_f32_dev(ap + 0, m.x);
  atomic_add_f32_dev(ap + 1, m.y);
  atomic_add_f32_dev(ap + 2, m.z);
  atomic_add_f32_dev(ap + 3, m.w);

  if (l == 0) wrel_out[e] = wrel;
}

// ---------------------------------------------------------------------------
// Kernel 3: out = agg @ nn_w.T + nn_b  via V_WMMA_F32_16X16X4_F32.
// Block = 128 threads = 4 waves; each wave computes one 16x16 tile
// (16 rows of `agg` x 16 output columns), K=64 unrolled in steps of 4.
//
// A 16x4 f32 layout (ISA 7.12.2): lanes 0-15 hold M=lane, K={0,1} in
// VGPR{0,1}; lanes 16-31 hold M=lane-16, K={2,3}.  -> per-lane float2 at
// agg[(row_base + (lane&15))*64 + k0 + 2*(lane>>4)].
// B 4x16 f32 (rows striped across lanes): same index pattern on nn_w since
// B[k][n] = nn_w[n][k].
// C/D 16x16 f32: VGPR v -> M = v + 8*(lane>>4), N = lane&15.
// ---------------------------------------------------------------------------
__global__ void cadi_gemm_wmma(const float* __restrict__ agg,
                               const float* __restrict__ nn_w,
                               const float* __restrict__ nn_b,
                               float* __restrict__ out) {
  const int lane = threadIdx.x & 31;
  const int wave = threadIdx.x >> 5;          // 0..3 -> column tile
  const int row_base = blockIdx.x * 16;
  const int col_base = wave * 16;
  const int l15 = lane & 15;
  const int khalf = (lane >> 4) * 2;          // 0 or 2

  const float* a_ptr = agg  + (size_t)(row_base + l15) * H_DIM + khalf;
  const float* b_ptr = nn_w + (size_t)(col_base + l15) * H_DIM + khalf;

  v8f c = {0.f, 0.f, 0.f, 0.f, 0.f, 0.f, 0.f, 0.f};
#pragma unroll
  for (int k0 = 0; k0 < H_DIM; k0 += 4) {
    v2f a = *(const v2f*)(a_ptr + k0);
    v2f b = *(const v2f*)(b_ptr + k0);
    // D = A x B + C  (exact fp32 WMMA)
    c = __builtin_amdgcn_wmma_f32_16x16x4_f32(
        /*neg_a=*/false, a, /*neg_b=*/false, b,
        /*c_mod=*/(short)0, c, /*reuse_a=*/false, /*reuse_b=*/false);
  }

  const float bias = nn_b[col_base + l15];
  const int m_base = row_base + (lane >> 4) * 8;
#pragma unroll
  for (int v = 0; v < 8; ++v) {
    out[(size_t)(m_base + v) * H_DIM + col_base + l15] = c[v] + bias;
  }
}

// ---------------------------------------------------------------------------
extern "C" void kernel_launch(void* const* d_in, const int* in_sizes, int n_in,
                              void* d_out, int out_size, void* d_ws, size_t ws_size,
                              hipStream_t stream) {
  const float* x         = (const float*)d_in[0];
  const int*   edge_idx  = (const int*)  d_in[1];
  const float* edge_attr = (const float*)d_in[2];
  const float* attn      = (const float*)d_in[3];
  const float* nn_w      = (const float*)d_in[4];
  const float* nn_b      = (const float*)d_in[5];
  const float* wr_w      = (const float*)d_in[6];
  const float* wr_b      = (const float*)d_in[7];
  const float* eps       = (const float*)d_in[8];

  const int N = in_sizes[0] / H_DIM;       // 50000
  const int E = in_sizes[1] / 2;           // 1600000
  const int* src = edge_idx;               // edge_index[0]
  const int* dst = edge_idx + E;           // edge_index[1]

  float* agg      = (float*)d_ws;          // N*64 floats scratch
  float* out_mat  = (float*)d_out;         // [N,64]
  float* wrel_out = (float*)d_out + (size_t)N * H_DIM;  // [E]

  // 1) agg = (1+eps)*x
  {
    const int n4 = (N * H_DIM) / 4;
    const int blk = 256;
    cadi_init<<<(n4 + blk - 1) / blk, blk, 0, stream>>>(x, eps, agg, n4);
  }
  // 2) edge messages + scatter add (16 threads per edge)
  {
    const long long threads = (long long)E * 16;
    const int blk = 256;
    const int grid = (int)((threads + blk - 1) / blk);
    cadi_edge<<<grid, blk, 0, stream>>>(x, src, dst, edge_attr, attn,
                                        wr_w, wr_b, agg, wrel_out, E);
  }
  // 3) out = agg @ nn_w.T + nn_b   (WMMA f32 16x16x4; N divisible by 16)
  {
    const int grid = N / 16;               // 3125 tiles of 16 rows
    cadi_gemm_wmma<<<grid, 128, 0, stream>>>(agg, nn_w, nn_b, out_mat);
  }
}